// MHF_SpectralConv_TFNO_29738353558079
// MI455X (gfx1250) — compile-verified
//
#include <hip/hip_runtime.h>
#include <math.h>

// ---------- CDNA5 WMMA f32 16x16x4 ----------
typedef float v2f __attribute__((ext_vector_type(2)));
typedef float v8f __attribute__((ext_vector_type(8)));

__device__ __forceinline__ v8f wmma4(v2f a, v2f b, v8f c) {
  // D = A(16x4) * B(4x16) + C, fp32, wave32
  return __builtin_amdgcn_wmma_f32_16x16x4_f32(false, a, false, b, (short)0, c, false, false);
}
__device__ __forceinline__ v8f vzero8() {
  v8f z = {0.f,0.f,0.f,0.f,0.f,0.f,0.f,0.f};
  return z;
}
__device__ __forceinline__ float2 cfma(float2 a, float2 b, float2 c) {
  return make_float2(fmaf(a.x, b.x, fmaf(-a.y, b.y, c.x)),
                     fmaf(a.x, b.y, fmaf( a.y, b.x, c.y)));
}

// Sizes
#define NB   8
#define NC   64
#define NH   256
#define NW   256
#define NR   32      // kept modes per axis (only r=32 survives the .set overwrites)
#define RANK 8

// ---------- 1) Exact-phase DFT basis tables (all phases are k*2pi/256) ----------
__global__ void init_basis(float2* __restrict__ Fw, float2* __restrict__ Fh,
                           float2* __restrict__ Eh, float* __restrict__ C2,
                           float* __restrict__ S2) {
  int i = blockIdx.x * blockDim.x + threadIdx.x;
  if (i >= 8192) return;
  const float STEP = 6.28318530717958647692f / 256.0f;
  const float invN = 1.0f / 65536.0f;   // norm='forward' on rfftn
  float s, c;
  { // Fw[w][n] = exp(-2pi i * v_n * w / 256)/65536, v_n = 48+n
    int w = i >> 5, n = i & 31;
    int p = ((48 + n) * w) & 255;
    __sincosf(STEP * (float)p, &s, &c);
    Fw[i] = make_float2(c * invN, -s * invN);
  }
  { // Fh[m][h] = exp(-2pi i * u_m * h / 256), u_m = m-16
    int m = i >> 8, h = i & 255;
    int p = ((m - 16) * h) & 255;
    __sincosf(STEP * (float)p, &s, &c);
    Fh[i] = make_float2(c, -s);
  }
  { // Eh[h][m] = exp(+2pi i * u_m * h / 256)
    int h = i >> 5, m = i & 31;
    int p = ((m - 16) * h) & 255;
    __sincosf(STEP * (float)p, &s, &c);
    Eh[i] = make_float2(c, s);
  }
  { // C2/S2[n][w]: Hermitian-doubled inverse w-basis (v strictly interior)
    int n = i >> 8, w = i & 255;
    int p = ((48 + n) * w) & 255;
    __sincosf(STEP * (float)p, &s, &c);
    C2[i] =  2.0f * c;
    S2[i] = -2.0f * s;
  }
}

// ---------- 2) K[a,q,m,n] = sum_{c,d} core[a,q,c,d] um1[m,c] um2[n,d] ----------
__global__ void kmix_kernel(const float2* __restrict__ core,
                            const float2* __restrict__ um1,
                            const float2* __restrict__ um2,
                            float2* __restrict__ Km) {
  int aq = blockIdx.x;                         // 0..63
  int mn = blockIdx.y * 256 + threadIdx.x;     // 0..1023
  int m = mn >> 5, n = mn & 31;
  const float2* cr = core + aq * 64;
  float2 acc = make_float2(0.f, 0.f);
  #pragma unroll
  for (int d = 0; d < 8; ++d) {
    float2 sd = make_float2(0.f, 0.f);
    #pragma unroll
    for (int c = 0; c < 8; ++c) sd = cfma(cr[c * 8 + d], um1[m * 8 + c], sd);
    acc = cfma(sd, um2[n * 8 + d], acc);
  }
  Km[aq * 1024 + mn] = acc;
}

// ---------- 3) Forward partial DFT: Xr[b,i,m,n] = Fh * (x * Fw) ----------
extern __shared__ float2 smem_f2[];

__global__ void __launch_bounds__(256) stageA(const float* __restrict__ x,
                                              const float2* __restrict__ Fw,
                                              const float2* __restrict__ Fh,
                                              float2* __restrict__ Xr) {
  float2* Tlds = smem_f2;                       // 256 x 32 complex (64KB)
  const int img  = blockIdx.x;                  // b*64 + i  (512 images)
  const float* xp = x + (size_t)img * (NH * NW);
  const int tid = threadIdx.x;
  const int wave = tid >> 5, lane = tid & 31;
  const int lm = lane & 15, half = lane >> 4;
  const int koff = half * 2;

  // ---- phase 1: T[h,n] = sum_w x[h,w] * Fw[w,n]  (M=256, N=32, K=256) ----
  v8f accR[2][2], accI[2][2];
  #pragma unroll
  for (int j = 0; j < 2; ++j)
    #pragma unroll
    for (int t = 0; t < 2; ++t) { accR[j][t] = vzero8(); accI[j][t] = vzero8(); }

  for (int k0 = 0; k0 < 256; k0 += 4) {
    const int kk = k0 + koff;
    v2f a[2];
    #pragma unroll
    for (int j = 0; j < 2; ++j) {
      int h = (wave * 2 + j) * 16 + lm;
      const float2 av = *(const float2*)(xp + h * NW + kk);   // x[h][kk], x[h][kk+1]
      v2f t; t.x = av.x; t.y = av.y; a[j] = t;
    }
    #pragma unroll
    for (int t = 0; t < 2; ++t) {
      int n = t * 16 + lm;
      float2 f0 = Fw[kk * 32 + n];
      float2 f1 = Fw[(kk + 1) * 32 + n];
      v2f bR = {f0.x, f1.x};
      v2f bI = {f0.y, f1.y};
      #pragma unroll
      for (int j = 0; j < 2; ++j) {
        accR[j][t] = wmma4(a[j], bR, accR[j][t]);
        accI[j][t] = wmma4(a[j], bI, accI[j][t]);
      }
    }
  }
  #pragma unroll
  for (int j = 0; j < 2; ++j)
    #pragma unroll
    for (int t = 0; t < 2; ++t)
      #pragma unroll
      for (int r = 0; r < 8; ++r) {
        int h = (wave * 2 + j) * 16 + r + half * 8;
        int n = t * 16 + lm;
        Tlds[h * 32 + n] = make_float2(accR[j][t][r], accI[j][t][r]);
      }
  __syncthreads();

  // ---- phase 2: Xr[m,n] = sum_h Fh[m,h] * T[h,n]  (M=32, N=32, K=256, cplx) ----
  const int tix = wave & 3, part = wave >> 2;   // 4 tiles x {Re,Im}
  const int mt = tix >> 1, nt = tix & 1;
  const int m = mt * 16 + lm, n = nt * 16 + lm;
  v8f acc = vzero8();
  for (int k0 = 0; k0 < 256; k0 += 4) {
    const int kk = k0 + koff;
    const float4 fa = *(const float4*)(&Fh[m * 256 + kk]);    // (re,im)(re,im)
    float2 t0 = Tlds[kk * 32 + n];
    float2 t1 = Tlds[(kk + 1) * 32 + n];
    v2f BR = {t0.x, t1.x};
    v2f BI = {t0.y, t1.y};
    v2f AR = {fa.x, fa.z};
    if (part == 0) {                // Re = AR*BR - AI*BI
      v2f AIn = {-fa.y, -fa.w};
      acc = wmma4(AR, BR, acc);
      acc = wmma4(AIn, BI, acc);
    } else {                        // Im = AR*BI + AI*BR
      v2f AI = {fa.y, fa.w};
      acc = wmma4(AR, BI, acc);
      acc = wmma4(AI, BR, acc);
    }
  }
  float* xout = (float*)(Xr + (size_t)img * (NR * NR));
  #pragma unroll
  for (int r = 0; r < 8; ++r) {
    int mm = mt * 16 + r + half * 8;
    xout[(mm * 32 + n) * 2 + part] = acc[r];
  }
}

// ---------- 4) Channel/mode mixing (rank-8; VALU is the right tool) ----------
__global__ void __launch_bounds__(256) chanmix(const float2* __restrict__ Xr,
                                               const float2* __restrict__ ui,
                                               const float2* __restrict__ uo,
                                               const float2* __restrict__ Km,
                                               float2* __restrict__ Yb) {
  __shared__ float2 uis[NC * RANK];
  __shared__ float2 uos[NC * RANK];
  const int tid = threadIdx.x;
  for (int idx = tid; idx < NC * RANK; idx += 256) { uis[idx] = ui[idx]; uos[idx] = uo[idx]; }
  __syncthreads();
  const int b = blockIdx.x;
  const int mn = blockIdx.y * 256 + tid;

  float2 Z[RANK];
  #pragma unroll
  for (int a = 0; a < RANK; ++a) Z[a] = make_float2(0.f, 0.f);
  for (int i = 0; i < NC; ++i) {
    float2 xv = Xr[((size_t)(b * NC + i)) * 1024 + mn];
    #pragma unroll
    for (int a = 0; a < RANK; ++a) Z[a] = cfma(xv, uis[i * RANK + a], Z[a]);
  }
  float2 P[RANK];
  #pragma unroll
  for (int q = 0; q < RANK; ++q) P[q] = make_float2(0.f, 0.f);
  #pragma unroll
  for (int a = 0; a < RANK; ++a) {
    #pragma unroll
    for (int q = 0; q < RANK; ++q)
      P[q] = cfma(Z[a], Km[(a * RANK + q) * 1024 + mn], P[q]);
  }
  for (int o = 0; o < NC; ++o) {
    float2 y = make_float2(0.f, 0.f);
    #pragma unroll
    for (int q = 0; q < RANK; ++q) y = cfma(uos[o * RANK + q], P[q], y);
    Yb[((size_t)(b * NC + o)) * 1024 + mn] = y;
  }
}

// ---------- 5) Inverse partial DFT + bias: y = Re{Eh*Y} x [2cos|-2sin] ----------
__global__ void __launch_bounds__(256) stageC(const float2* __restrict__ Yb,
                                              const float2* __restrict__ Eh,
                                              const float*  __restrict__ C2,
                                              const float*  __restrict__ S2,
                                              const float*  __restrict__ bias,
                                              float* __restrict__ out) {
  float2* Ylds = smem_f2;            // 32x32 complex (8KB)
  float2* T2   = smem_f2 + 1024;     // 256x32 complex (64KB)
  const int img = blockIdx.x;        // b*64 + o
  const int o = img & 63;
  const int tid = threadIdx.x;
  const int wave = tid >> 5, lane = tid & 31;
  const int lm = lane & 15, half = lane >> 4;
  const int koff = half * 2;

  for (int idx = tid; idx < 1024; idx += 256) Ylds[idx] = Yb[(size_t)img * 1024 + idx];
  __syncthreads();

  // ---- C1: T2[h,n] = sum_m Eh[h,m] * Y[m,n]  (M=256, N=32, K=32, cplx) ----
  v8f aR[2][2], aI[2][2];
  #pragma unroll
  for (int j = 0; j < 2; ++j)
    #pragma unroll
    for (int t = 0; t < 2; ++t) { aR[j][t] = vzero8(); aI[j][t] = vzero8(); }

  #pragma unroll
  for (int k0 = 0; k0 < 32; k0 += 4) {
    const int kk = k0 + koff;
    v2f AR[2], AIp[2], AIn[2];
    #pragma unroll
    for (int j = 0; j < 2; ++j) {
      int h = (wave * 2 + j) * 16 + lm;
      const float4 fa = *(const float4*)(&Eh[h * 32 + kk]);
      v2f r = {fa.x, fa.z};   AR[j]  = r;
      v2f ip = {fa.y, fa.w};  AIp[j] = ip;
      v2f in = {-fa.y, -fa.w}; AIn[j] = in;
    }
    #pragma unroll
    for (int t = 0; t < 2; ++t) {
      int n = t * 16 + lm;
      float2 t0 = Ylds[kk * 32 + n];
      float2 t1 = Ylds[(kk + 1) * 32 + n];
      v2f BR = {t0.x, t1.x};
      v2f BI = {t0.y, t1.y};
      #pragma unroll
      for (int j = 0; j < 2; ++j) {
        aR[j][t] = wmma4(AR[j],  BR, aR[j][t]);
        aR[j][t] = wmma4(AIn[j], BI, aR[j][t]);
        aI[j][t] = wmma4(AR[j],  BI, aI[j][t]);
        aI[j][t] = wmma4(AIp[j], BR, aI[j][t]);
      }
    }
  }
  #pragma unroll
  for (int j = 0; j < 2; ++j)
    #pragma unroll
    for (int t = 0; t < 2; ++t)
      #pragma unroll
      for (int r = 0; r < 8; ++r) {
        int h = (wave * 2 + j) * 16 + r + half * 8;
        int n = t * 16 + lm;
        T2[h * 32 + n] = make_float2(aR[j][t][r], aI[j][t][r]);
      }
  __syncthreads();

  // ---- C2: y[h,w] = sum_n ReT2*2cos + ImT2*(-2sin) + bias  (M=256,N=256,K=32) ----
  const float bo = bias[o];
  float* op = out + (size_t)img * (NH * NW);
  for (int tix = wave; tix < 256; tix += 8) {
    const int ht = tix >> 4, wt = tix & 15;
    const int h = ht * 16 + lm, w = wt * 16 + lm;
    v8f acc = vzero8();
    #pragma unroll
    for (int k0 = 0; k0 < 32; k0 += 4) {
      const int kk = k0 + koff;
      const float4 tt = *(const float4*)(&T2[h * 32 + kk]);   // (re,im)(re,im)
      v2f AR = {tt.x, tt.z};
      v2f AI = {tt.y, tt.w};
      v2f Bc = {C2[kk * 256 + w], C2[(kk + 1) * 256 + w]};
      v2f Bs = {S2[kk * 256 + w], S2[(kk + 1) * 256 + w]};
      acc = wmma4(AR, Bc, acc);
      acc = wmma4(AI, Bs, acc);
    }
    #pragma unroll
    for (int r = 0; r < 8; ++r) {
      int hh = ht * 16 + r + half * 8;
      op[hh * 256 + w] = acc[r] + bo;
    }
  }
}

// ---------- launch ----------
extern "C" void kernel_launch(void* const* d_in, const int* in_sizes, int n_in,
                              void* d_out, int out_size, void* d_ws, size_t ws_size,
                              hipStream_t stream) {
  (void)in_sizes; (void)n_in; (void)out_size; (void)ws_size;
  const float*  x    = (const float*)d_in[0];
  const float*  bias = (const float*)d_in[1];
  // only the r=32 branch survives the nested .set() overwrites
  const float2* core = (const float2*)d_in[17];
  const float2* ui   = (const float2*)d_in[18];
  const float2* uo   = (const float2*)d_in[19];
  const float2* um1  = (const float2*)d_in[20];
  const float2* um2  = (const float2*)d_in[21];

  float2* wsc = (float2*)d_ws;
  float2* Fw = wsc;                         //  8192 cplx
  float2* Fh = wsc + 8192;                  //  8192 cplx
  float2* Eh = wsc + 16384;                 //  8192 cplx
  float2* Km = wsc + 24576;                 // 65536 cplx
  float2* Xr = wsc + 90112;                 // 8*64*1024 cplx
  float2* Yb = wsc + 614400;                // 8*64*1024 cplx
  float*  C2 = (float*)(wsc + 1138688);     //  8192 f32
  float*  S2 = C2 + 8192;                   //  8192 f32

  init_basis<<<32, 256, 0, stream>>>(Fw, Fh, Eh, C2, S2);
  kmix_kernel<<<dim3(64, 4), 256, 0, stream>>>(core, um1, um2, Km);
  stageA<<<NB * NC, 256, 256 * 32 * sizeof(float2), stream>>>(x, Fw, Fh, Xr);
  chanmix<<<dim3(NB, 4), 256, 0, stream>>>(Xr, ui, uo, Km, Yb);
  stageC<<<NB * NC, 256, (1024 + 256 * 32) * sizeof(float2), stream>>>(
      Yb, Eh, C2, S2, bias, (float*)d_out);
}